// ResGCN5_58128087384885
// MI455X (gfx1250) — compile-verified
//
#include <hip/hip_runtime.h>
#include <math.h>

#define NNODES 50000
#define NEDGES 1600000
#define NFEAT  512
#define NHID   256
#define NCLASS 40

typedef float v2f __attribute__((ext_vector_type(2)));
typedef float v8f __attribute__((ext_vector_type(8)));

// ---------------------------------------------------------------- utilities

__global__ void zero_i32(int* __restrict__ p, int n) {
    int i = blockIdx.x * blockDim.x + threadIdx.x;
    if (i < n) p[i] = 0;
}

__global__ void count_dst(const int* __restrict__ dst, int* __restrict__ counts, int E) {
    int e = blockIdx.x * blockDim.x + threadIdx.x;
    if (e < E) atomicAdd(&counts[dst[e]], 1);
}

// single-block chunked Hillis-Steele exclusive scan: row_ptr[0..n] from counts[0..n-1]
__global__ void scan_counts(const int* __restrict__ counts, int* __restrict__ row_ptr, int n) {
    __shared__ int s[1024];
    __shared__ int carry;
    int tid = threadIdx.x;
    if (tid == 0) carry = 0;
    __syncthreads();
    for (int base = 0; base < n; base += 1024) {
        int i = base + tid;
        int v = (i < n) ? counts[i] : 0;
        s[tid] = v;
        __syncthreads();
        for (int off = 1; off < 1024; off <<= 1) {
            int t = (tid >= off) ? s[tid - off] : 0;
            __syncthreads();
            s[tid] += t;
            __syncthreads();
        }
        if (i < n) row_ptr[i] = carry + s[tid] - v;   // exclusive
        __syncthreads();
        if (tid == 0) carry += s[1023];
        __syncthreads();
    }
    if (threadIdx.x == 0) row_ptr[n] = carry;
}

__global__ void scatter_edges(const int* __restrict__ src, const int* __restrict__ dst,
                              const float* __restrict__ vals,
                              const int* __restrict__ row_ptr, int* __restrict__ fill,
                              int* __restrict__ ssrc, float* __restrict__ sval, int E) {
    int e = blockIdx.x * blockDim.x + threadIdx.x;
    if (e < E) {
        int d = dst[e];
        int pos = row_ptr[d] + atomicAdd(&fill[d], 1);
        ssrc[pos] = src[e];
        sval[pos] = vals[e];
    }
}

// zero-pad W5 [1024,40] -> [1024,64]
__global__ void pad_W5(const float* __restrict__ W5, float* __restrict__ W5p) {
    int i = blockIdx.x * blockDim.x + threadIdx.x;
    if (i < 1024 * 64) {
        int r = i >> 6, c = i & 63;
        W5p[i] = (c < NCLASS) ? W5[r * NCLASS + c] : 0.0f;
    }
}

// ------------------------------------------------------- f32 WMMA GEMM
// C[M,N] = A[M,K] @ B[K,N] (+bias), row-major.
// One wave32 computes a 16x64 macro-tile: one A fragment feeds 4
// V_WMMA_F32_16X16X4_F32 tiles per K-step. N must be a multiple of 64
// (callers pad B/C); M must be a multiple of 16.
// Fragment layouts (f32 16x16x4):
//   A: lane l holds A[m][k + 2*(l>>4) + r] in vgpr r   (one float2 load)
//   B: lane l holds B[k + 2*(l>>4) + r][n] in vgpr r
//   C/D: vgpr r, lane l -> row = r + 8*(l>>4), col = l&15
__global__ void gemm_wmma_f32(const float* __restrict__ A, int lda,
                              const float* __restrict__ B, int ldb,
                              float* __restrict__ C, int ldc,
                              const float* __restrict__ bias,
                              int K, int tilesN, int tiles) {
    int w = (blockIdx.x * blockDim.x + threadIdx.x) >> 5;
    if (w >= tiles) return;
    int lane = threadIdx.x & 31;
    int tm = w / tilesN, tn = w % tilesN;
    int half = lane >> 4;
    int l15  = lane & 15;
    int mrow  = tm * 16 + l15;
    int ncol0 = tn * 64 + l15;

    const float* Ap = A + (size_t)mrow * lda + 2 * half;
    const float* Bp = B + (size_t)(2 * half) * ldb + ncol0;

    v8f acc0 = {}, acc1 = {}, acc2 = {}, acc3 = {};
#pragma unroll 2
    for (int k = 0; k < K; k += 4) {
        v2f a = *(const v2f*)Ap;
        v2f b0, b1, b2, b3;
        b0.x = Bp[0];   b0.y = Bp[ldb];
        b1.x = Bp[16];  b1.y = Bp[ldb + 16];
        b2.x = Bp[32];  b2.y = Bp[ldb + 32];
        b3.x = Bp[48];  b3.y = Bp[ldb + 48];
        acc0 = __builtin_amdgcn_wmma_f32_16x16x4_f32(false, a, false, b0, (short)0, acc0, false, false);
        acc1 = __builtin_amdgcn_wmma_f32_16x16x4_f32(false, a, false, b1, (short)0, acc1, false, false);
        acc2 = __builtin_amdgcn_wmma_f32_16x16x4_f32(false, a, false, b2, (short)0, acc2, false, false);
        acc3 = __builtin_amdgcn_wmma_f32_16x16x4_f32(false, a, false, b3, (short)0, acc3, false, false);
        Ap += 4;
        Bp += (size_t)4 * ldb;
    }

    float bv0 = 0.f, bv1 = 0.f, bv2 = 0.f, bv3 = 0.f;
    if (bias) {
        bv0 = bias[ncol0];
        bv1 = bias[ncol0 + 16];
        bv2 = bias[ncol0 + 32];
        bv3 = bias[ncol0 + 48];
    }
#pragma unroll
    for (int r = 0; r < 8; ++r) {
        size_t rowoff = (size_t)(tm * 16 + r + 8 * half) * ldc;
        C[rowoff + ncol0]      = acc0[r] + bv0;
        C[rowoff + ncol0 + 16] = acc1[r] + bv1;
        C[rowoff + ncol0 + 32] = acc2[r] + bv2;
        C[rowoff + ncol0 + 48] = acc3[r] + bv3;
    }
}

// --------------------------------------------------- CSR SpMM, D = 256
// out[row, :] = relu(sum_e val*H[src,:] + bias) + res[row,:]
__global__ void spmm_csr_d256(const int* __restrict__ row_ptr,
                              const int* __restrict__ ssrc,
                              const float* __restrict__ sval,
                              const float* __restrict__ H,
                              const float* __restrict__ bias,
                              const float* __restrict__ res, int ldres,
                              float* __restrict__ out, int ldo,
                              int nrows) {
    int w = (blockIdx.x * blockDim.x + threadIdx.x) >> 5;
    if (w >= nrows) return;
    int lane = threadIdx.x & 31;
    int c0 = lane * 8;
    float acc[8];
#pragma unroll
    for (int i = 0; i < 8; ++i) acc[i] = 0.0f;
    int beg = row_ptr[w], end = row_ptr[w + 1];
    for (int j = beg; j < end; ++j) {
        int s = ssrc[j];
        float v = sval[j];
        const float4* hp = (const float4*)(H + (size_t)s * 256 + c0);
        float4 h0 = hp[0], h1 = hp[1];
        acc[0] = fmaf(v, h0.x, acc[0]); acc[1] = fmaf(v, h0.y, acc[1]);
        acc[2] = fmaf(v, h0.z, acc[2]); acc[3] = fmaf(v, h0.w, acc[3]);
        acc[4] = fmaf(v, h1.x, acc[4]); acc[5] = fmaf(v, h1.y, acc[5]);
        acc[6] = fmaf(v, h1.z, acc[6]); acc[7] = fmaf(v, h1.w, acc[7]);
    }
    const float* rrow = res + (size_t)w * ldres + c0;
    float t[8];
#pragma unroll
    for (int i = 0; i < 8; ++i) {
        float u = acc[i] + bias[c0 + i];
        u = fmaxf(u, 0.0f);          // relu
        t[i] = u + rrow[i];          // residual (may alias out; per-thread RAW is safe)
    }
    float4* op = (float4*)(out + (size_t)w * ldo + c0);
    op[0] = make_float4(t[0], t[1], t[2], t[3]);
    op[1] = make_float4(t[4], t[5], t[6], t[7]);
}

// --------------------------------------------------- CSR SpMM, D = 40 (final layer)
__global__ void spmm_csr_d40(const int* __restrict__ row_ptr,
                             const int* __restrict__ ssrc,
                             const float* __restrict__ sval,
                             const float* __restrict__ H, int ldh,
                             const float* __restrict__ bias,
                             float* __restrict__ out, int nrows) {
    int w = (blockIdx.x * blockDim.x + threadIdx.x) >> 5;
    if (w >= nrows) return;
    int lane = threadIdx.x & 31;
    int c = lane * 2;
    if (c >= NCLASS) return;     // no cross-lane ops below; early-out is safe
    float a0 = 0.0f, a1 = 0.0f;
    int beg = row_ptr[w], end = row_ptr[w + 1];
    for (int j = beg; j < end; ++j) {
        int s = ssrc[j];
        float v = sval[j];
        const float2* hp = (const float2*)(H + (size_t)s * ldh + c);
        float2 h = *hp;
        a0 = fmaf(v, h.x, a0);
        a1 = fmaf(v, h.y, a1);
    }
    out[(size_t)w * NCLASS + c]     = a0 + bias[c];
    out[(size_t)w * NCLASS + c + 1] = a1 + bias[c + 1];
}

// --------------------------------------------------- log_softmax over 40 classes
__global__ void log_softmax40(float* __restrict__ out, int nrows) {
    int w = (blockIdx.x * blockDim.x + threadIdx.x) >> 5;
    if (w >= nrows) return;
    int lane = threadIdx.x & 31;
    int c = lane * 2;
    float v0 = (c     < NCLASS) ? out[(size_t)w * NCLASS + c]     : -INFINITY;
    float v1 = (c + 1 < NCLASS) ? out[(size_t)w * NCLASS + c + 1] : -INFINITY;
    float m = fmaxf(v0, v1);
#pragma unroll
    for (int off = 16; off > 0; off >>= 1) m = fmaxf(m, __shfl_xor(m, off, 32));
    float e = ((c < NCLASS) ? expf(v0 - m) : 0.0f) + ((c + 1 < NCLASS) ? expf(v1 - m) : 0.0f);
#pragma unroll
    for (int off = 16; off > 0; off >>= 1) e += __shfl_xor(e, off, 32);
    float lse = m + logf(e);
    if (c     < NCLASS) out[(size_t)w * NCLASS + c]     = v0 - lse;
    if (c + 1 < NCLASS) out[(size_t)w * NCLASS + c + 1] = v1 - lse;
}

// ---------------------------------------------------------------- launcher

static inline size_t align256(size_t x) { return (x + 255) & ~(size_t)255; }

extern "C" void kernel_launch(void* const* d_in, const int* in_sizes, int n_in,
                              void* d_out, int out_size, void* d_ws, size_t ws_size,
                              hipStream_t stream) {
    const float* x        = (const float*)d_in[0];
    const int*   edge_src = (const int*)  d_in[1];
    const int*   edge_dst = (const int*)  d_in[2];
    const float* edge_val = (const float*)d_in[3];
    const float* W0 = (const float*)d_in[4];  const float* b0 = (const float*)d_in[5];
    const float* W1 = (const float*)d_in[6];  const float* b1 = (const float*)d_in[7];
    const float* W2 = (const float*)d_in[8];  const float* b2 = (const float*)d_in[9];
    const float* W3 = (const float*)d_in[10]; const float* b3 = (const float*)d_in[11];
    const float* W5 = (const float*)d_in[12]; const float* b5 = (const float*)d_in[13];
    float* out = (float*)d_out;

    // workspace carve-up
    char* p = (char*)d_ws;
    float* xc      = (float*)p; p += align256((size_t)NNODES * 1024 * 4);  // concat buffer
    float* h_tmp   = (float*)p; p += align256((size_t)NNODES * 256 * 4);   // x@W per layer
    float* h5      = (float*)p; p += align256((size_t)NNODES * 64 * 4);    // padded N=40->64
    float* W5p     = (float*)p; p += align256((size_t)1024 * 64 * 4);      // padded W5
    int*   row_ptr = (int*)p;   p += align256((size_t)(NNODES + 1) * 4);
    int*   counts  = (int*)p;   p += align256((size_t)NNODES * 4);
    int*   fill    = (int*)p;   p += align256((size_t)NNODES * 4);
    int*   ssrc    = (int*)p;   p += align256((size_t)NEDGES * 4);
    float* sval    = (float*)p; p += align256((size_t)NEDGES * 4);

    // column slices of xc = concat(x4, x3, x2, x1)
    float* x4p = xc + 0;
    float* x3p = xc + 256;
    float* x2p = xc + 512;
    float* x1p = xc + 768;   // z is staged here, then overwritten by x1

    const int TB = 256;
    int blkN = (NNODES + TB - 1) / TB;
    int blkE = (NEDGES + TB - 1) / TB;
    int blkRows = (NNODES * 32 + TB - 1) / TB;   // one wave32 per row

    // ---- build dst-CSR (once per call)
    zero_i32<<<blkN, TB, 0, stream>>>(counts, NNODES);
    zero_i32<<<blkN, TB, 0, stream>>>(fill, NNODES);
    count_dst<<<blkE, TB, 0, stream>>>(edge_dst, counts, NEDGES);
    scan_counts<<<1, 1024, 0, stream>>>(counts, row_ptr, NNODES);
    scatter_edges<<<blkE, TB, 0, stream>>>(edge_src, edge_dst, edge_val,
                                           row_ptr, fill, ssrc, sval, NEDGES);
    pad_W5<<<(1024 * 64 + TB - 1) / TB, TB, 0, stream>>>(W5, W5p);

    auto gemm = [&](const float* A, int lda, const float* B, int ldb,
                    float* C, int ldc, const float* bias, int Npad, int K) {
        int tilesM = NNODES / 16;
        int tilesN = Npad / 64;            // 16x64 macro-tile per wave
        int tiles = tilesM * tilesN;
        int blocks = (tiles + 7) / 8;      // 8 waves per 256-thread block
        gemm_wmma_f32<<<blocks, TB, 0, stream>>>(A, lda, B, ldb, C, ldc, bias,
                                                 K, tilesN, tiles);
    };

    // z = x@W0 + b0  (staged into x1 slot)
    gemm(x, NFEAT, W0, NHID, x1p, 1024, b0, NHID, NFEAT);
    // x1 = relu(spmm(x@W1) + b1) + z
    gemm(x, NFEAT, W1, NHID, h_tmp, NHID, nullptr, NHID, NFEAT);
    spmm_csr_d256<<<blkRows, TB, 0, stream>>>(row_ptr, ssrc, sval, h_tmp, b1,
                                              x1p, 1024, x1p, 1024, NNODES);
    // x2 = relu(spmm(x1@W2) + b2) + x1
    gemm(x1p, 1024, W2, NHID, h_tmp, NHID, nullptr, NHID, NHID);
    spmm_csr_d256<<<blkRows, TB, 0, stream>>>(row_ptr, ssrc, sval, h_tmp, b2,
                                              x1p, 1024, x2p, 1024, NNODES);
    // x3 = relu(spmm(x2@W2) + b2) + x2
    gemm(x2p, 1024, W2, NHID, h_tmp, NHID, nullptr, NHID, NHID);
    spmm_csr_d256<<<blkRows, TB, 0, stream>>>(row_ptr, ssrc, sval, h_tmp, b2,
                                              x2p, 1024, x3p, 1024, NNODES);
    // x4 = relu(spmm(x3@W3) + b3) + x3
    gemm(x3p, 1024, W3, NHID, h_tmp, NHID, nullptr, NHID, NHID);
    spmm_csr_d256<<<blkRows, TB, 0, stream>>>(row_ptr, ssrc, sval, h_tmp, b3,
                                              x3p, 1024, x4p, 1024, NNODES);
    // x5 = spmm(xc@W5) + b5 ; then log_softmax
    gemm(xc, 1024, W5p, 64, h5, 64, nullptr, 64, 4 * NHID);
    spmm_csr_d40<<<blkRows, TB, 0, stream>>>(row_ptr, ssrc, sval, h5, 64, b5,
                                             out, NNODES);
    log_softmax40<<<blkRows, TB, 0, stream>>>(out, NNODES);
    (void)in_sizes; (void)n_in; (void)out_size; (void)ws_size;
}